// DifferentiableAggregation_test_6330781794349
// MI455X (gfx1250) — compile-verified
//
#include <hip/hip_runtime.h>

#define MAXB 8192
#define ACC_BLOCK 256

typedef float v2f __attribute__((ext_vector_type(2)));
typedef float v8f __attribute__((ext_vector_type(8)));

__device__ __forceinline__ float log_sigmoid_eps(float x) {
  float s = 1.0f / (1.0f + __expf(-x));
  return logf(s + 1e-10f);
}

// ---------------- privatized segment accumulation (LDS atomics) ----------------
__global__ void __launch_bounds__(ACC_BLOCK)
seg_accum_lds(const float* __restrict__ L, const int* __restrict__ idx,
              float* __restrict__ partial, int total, int B) {
  __shared__ float acc[2 * MAXB];   // 64 KB: s1 at even slots, s0 at odd slots
  const int nslots = 2 * B;
  for (int i = threadIdx.x; i < nslots; i += blockDim.x) acc[i] = 0.0f;
  __syncthreads();

  const int tid = blockIdx.x * blockDim.x + threadIdx.x;
  const int stride = gridDim.x * blockDim.x;
  const int nchunk = total >> 2;                       // 4 rows (48B) per chunk
  const float4* __restrict__ L4 = (const float4*)L;
  const int4*   __restrict__ I4 = (const int4*)idx;

  for (int j = tid; j < nchunk; j += stride) {
    float4 x = L4[3 * j + 0];
    float4 y = L4[3 * j + 1];
    float4 z = L4[3 * j + 2];
    int4 id = I4[j];
    // rows: (x.x,x.y,x.z) (x.w,y.x,y.y) (y.z,y.w,z.x) (z.y,z.z,z.w)
    atomicAdd(&acc[2 * id.x],     x.y + x.z);
    atomicAdd(&acc[2 * id.x + 1], x.x);
    atomicAdd(&acc[2 * id.y],     y.x + y.y);
    atomicAdd(&acc[2 * id.y + 1], x.w);
    atomicAdd(&acc[2 * id.z],     y.w + z.x);
    atomicAdd(&acc[2 * id.z + 1], y.z);
    atomicAdd(&acc[2 * id.w],     z.z + z.w);
    atomicAdd(&acc[2 * id.w + 1], z.y);
  }
  // scalar tail
  for (int i = (nchunk << 2) + tid; i < total; i += stride) {
    float a = L[3 * i], b1 = L[3 * i + 1], c1 = L[3 * i + 2];
    int b = idx[i];
    atomicAdd(&acc[2 * b],     b1 + c1);
    atomicAdd(&acc[2 * b + 1], a);
  }
  __syncthreads();

  // flush private accumulator to this block's partial buffer (coalesced b128)
  float4* __restrict__ dst4 = (float4*)(partial + (size_t)blockIdx.x * nslots);
  const float4* __restrict__ src4 = (const float4*)acc;
  for (int i = threadIdx.x; i < (nslots >> 2); i += blockDim.x) dst4[i] = src4[i];
}

// ---- cross-block reduction as WMMA: D = ones(16x4) * P(4x16) + C, looped over G ----
// Each wave reduces 16 output slots; column n = lane%16, k-rows come from g..g+3.
__global__ void __launch_bounds__(256)
reduce_wmma(const float* __restrict__ partial, float* __restrict__ out,
            int B, int G) {
  const int tid  = blockIdx.x * blockDim.x + threadIdx.x;
  const int wave = tid >> 5;
  const int lane = tid & 31;
  const int nslots = 2 * B;
  const int base = wave * 16;           // wave-uniform
  if (base >= nslots) return;           // whole wave exits together

  v2f a; a[0] = 1.0f; a[1] = 1.0f;      // ones A-matrix fragment
  v8f c = {};
  const int colLane = lane & 15;
  const int krow = (lane >> 4) * 2;     // lanes 0-15 -> k{0,1}, lanes 16-31 -> k{2,3}
  const float* colptr = partial + base + colLane;

  for (int g = 0; g < G; g += 4) {
    v2f b;
    b[0] = colptr[(size_t)(g + krow)     * nslots];
    b[1] = colptr[(size_t)(g + krow + 1) * nslots];
    c = __builtin_amdgcn_wmma_f32_16x16x4_f32(false, a, false, b,
                                              (short)0, c, false, false);
  }

  if (lane < 16) {
    int slot = base + lane;
    float s = c[0];                     // row M=0 of D = column sum
    float bias = (slot & 1) ? 5.0f : 1.0f;   // even slot: s1 path, odd: s0 path
    out[slot] = log_sigmoid_eps(10.0f * (bias - s));
  }
}

// ---------------- fallback path (workspace too small): direct global atomics ----------------
__global__ void zero_f32(float* __restrict__ p, int n) {
  int i = blockIdx.x * blockDim.x + threadIdx.x;
  if (i < n) p[i] = 0.0f;
}

__global__ void seg_accum_direct(const float* __restrict__ L, const int* __restrict__ idx,
                                 float* __restrict__ acc, int total) {
  int tid = blockIdx.x * blockDim.x + threadIdx.x;
  int stride = gridDim.x * blockDim.x;
  for (int i = tid; i < total; i += stride) {
    float a = L[3 * i], b1 = L[3 * i + 1], c1 = L[3 * i + 2];
    int b = idx[i];
    atomicAdd(&acc[2 * b],     b1 + c1);
    atomicAdd(&acc[2 * b + 1], a);
  }
}

__global__ void transform_inplace(float* __restrict__ out, int B) {
  int b = blockIdx.x * blockDim.x + threadIdx.x;
  if (b < B) {
    float s1 = out[2 * b];
    float s0 = out[2 * b + 1];
    out[2 * b]     = log_sigmoid_eps(10.0f * (1.0f - s1));
    out[2 * b + 1] = log_sigmoid_eps(10.0f * (5.0f - s0));
  }
}

extern "C" void kernel_launch(void* const* d_in, const int* in_sizes, int n_in,
                              void* d_out, int out_size, void* d_ws, size_t ws_size,
                              hipStream_t stream) {
  const float* L  = (const float*)d_in[0];
  const int* idx  = (const int*)d_in[1];
  // d_in[2] is batch_size as a device scalar; B is recovered from out_size.
  float* out = (float*)d_out;
  const int B = out_size / 2;
  const int total = in_sizes[0] / 3;

  const size_t per = (size_t)(2 * B) * sizeof(float);
  long long gmax = per ? (long long)(ws_size / per) : 0;
  int G = (int)(gmax > 512 ? 512 : gmax);
  G &= ~3;  // WMMA reduction consumes partials 4 at a time

  if (B > 0 && B <= MAXB && G >= 4 && (2 * B) % 16 == 0) {
    float* partial = (float*)d_ws;
    seg_accum_lds<<<G, ACC_BLOCK, 0, stream>>>(L, idx, partial, total, B);
    const int waves = (2 * B) / 16;
    const int threads = waves * 32;
    const int tpb = 256;
    reduce_wmma<<<(threads + tpb - 1) / tpb, tpb, 0, stream>>>(partial, out, B, G);
  } else if (B > 0) {
    const int n = 2 * B;
    zero_f32<<<(n + 255) / 256, 256, 0, stream>>>(out, n);
    seg_accum_direct<<<1024, 256, 0, stream>>>(L, idx, out, total);
    transform_inplace<<<(B + 255) / 256, 256, 0, stream>>>(out, B);
  }
}